// CombineModel_18846316494851
// MI455X (gfx1250) — compile-verified
//
#include <hip/hip_runtime.h>
#include <math.h>

#define N_NODES 8192
#define N_EDGES 262144
#define EMBED   128
#define EDGE_DIM 16
#define R_REL   8

typedef __attribute__((ext_vector_type(2))) float v2f;
typedef __attribute__((ext_vector_type(8))) float v8f;

// ---- monotone float<->uint ordering keys (for atomic max on floats) ----
__device__ __forceinline__ unsigned int fkey(float f) {
    unsigned int b = __float_as_uint(f);
    return (b & 0x80000000u) ? ~b : (b | 0x80000000u);
}
__device__ __forceinline__ float funkey(unsigned int u) {
    return __uint_as_float((u & 0x80000000u) ? (u & 0x7FFFFFFFu) : ~u);
}
#define KEY_NEG_INF 0x007FFFFFu   // fkey(-inf)

// ---- embedding gather: X0 = emb[x], also emit embedding_x output ----
__global__ void k_embed(const int* __restrict__ x, const float* __restrict__ emb,
                        float* __restrict__ X0, float* __restrict__ outEmb) {
    int i = blockIdx.x * blockDim.x + threadIdx.x;
    if (i >= N_NODES * EMBED) return;
    int n = i / EMBED, c = i % EMBED;
    float v = emb[x[n] * EMBED + c];
    X0[i] = v;
    outEmb[i] = v;
}

// ---- lev[d] = sum_j le[d][j] * evec[j]  (le is EDGE_DIM x out_dim) ----
__global__ void k_lev(const float* __restrict__ le, const float* __restrict__ evec,
                      float* __restrict__ lev, int out_dim) {
    int d = threadIdx.x;
    if (d < EDGE_DIM) {
        float s = 0.f;
        for (int j = 0; j < out_dim; ++j) s += le[d * out_dim + j] * evec[j];
        lev[d] = s;
    }
}

// ---- per-relation feature transform Y[r] = Xin @ W[r] via f32 WMMA ----
// grid: (N/16 row tiles, OUT/16 col tiles, R relations), 32 threads (1 wave)
// Dims are compile-time so the K loop is fully unrolled straight-line WMMA.
template <int IN, int OUT>
__global__ __launch_bounds__(32)
void k_gemm_x_w(const float* __restrict__ X, const float* __restrict__ W,
                float* __restrict__ Y) {
    const int lane = threadIdx.x;
    const int m    = lane & 15;     // row (A) / col (B) within tile
    const int kh   = lane >> 4;     // K half-select (0 -> K 0,1 ; 1 -> K 2,3)
    const int row0 = blockIdx.x * 16;
    const int n0   = blockIdx.y * 16;
    const int r    = blockIdx.z;
    const float* Wr = W + (size_t)r * IN * OUT;

    v8f c = {0.f, 0.f, 0.f, 0.f, 0.f, 0.f, 0.f, 0.f};
#pragma unroll
    for (int k = 0; k < IN; k += 4) {
        const int ka = k + kh * 2;
        v2f a = *(const v2f*)&X[(size_t)(row0 + m) * IN + ka];  // contiguous pair
        v2f b;
        b.x = Wr[(size_t)ka       * OUT + n0 + m];
        b.y = Wr[(size_t)(ka + 1) * OUT + n0 + m];
        c = __builtin_amdgcn_wmma_f32_16x16x4_f32(false, a, false, b,
                                                  (short)0, c, false, false);
    }
    float* Yr = Y + (size_t)r * N_NODES * OUT;
    const int mbase = kh * 8;
#pragma unroll
    for (int v = 0; v < 8; ++v)
        Yr[(size_t)(row0 + mbase + v) * OUT + n0 + m] = c[v];
}

// ---- per-(relation,node) attention scalars: AI = Y.q, AJ = Y.k ----
__global__ void k_scores(const float* __restrict__ Y, const float* __restrict__ q,
                         const float* __restrict__ kk, float* __restrict__ AI,
                         float* __restrict__ AJ, int out_dim) {
    int i = blockIdx.x * blockDim.x + threadIdx.x;      // r*N + node
    if (i >= R_REL * N_NODES) return;
    const float* y = Y + (size_t)i * out_dim;
    float si = 0.f, sj = 0.f;
    for (int c = 0; c < out_dim; ++c) { float v = y[c]; si += v * q[c]; sj += v * kk[c]; }
    AI[i] = si; AJ[i] = sj;
}

// ---- init accumulators ----
__global__ void k_init(unsigned int* __restrict__ amax, float* __restrict__ denom,
                       float* __restrict__ acc, int accN) {
    int i = blockIdx.x * blockDim.x + threadIdx.x;
    if (i < N_NODES) { amax[i] = KEY_NEG_INF; denom[i] = 0.f; }
    if (i < accN)    acc[i] = 0.f;
}

// ---- pass 1: raw logit + leaky_relu(0.2), atomic max per dst ----
__global__ void k_alpha(const int* __restrict__ src, const int* __restrict__ dst,
                        const int* __restrict__ etype, const float* __restrict__ eattr,
                        const float* __restrict__ lev, const float* __restrict__ AI,
                        const float* __restrict__ AJ, float* __restrict__ WE,
                        unsigned int* __restrict__ amax) {
    int e = blockIdx.x * blockDim.x + threadIdx.x;
    if (e >= N_EDGES) return;
    const int s = src[e], d = dst[e], r = etype[e];
    float ea = 0.f;
    const float* ep = eattr + (size_t)e * EDGE_DIM;
#pragma unroll
    for (int j = 0; j < EDGE_DIM; ++j) ea += ep[j] * lev[j];
    float a = AI[r * N_NODES + d] + AJ[r * N_NODES + s] + ea;
    a = (a > 0.f) ? a : 0.2f * a;
    WE[e] = a;
    atomicMax(&amax[d], fkey(a));
}

// ---- pass 2: exp(alpha - max), atomic sum per dst ----
__global__ void k_expsum(float* __restrict__ WE, const unsigned int* __restrict__ amax,
                         const int* __restrict__ dst, float* __restrict__ denom) {
    int e = blockIdx.x * blockDim.x + threadIdx.x;
    if (e >= N_EDGES) return;
    const int d = dst[e];
    float w = __expf(WE[e] - funkey(amax[d]));
    WE[e] = w;
    atomicAdd(&denom[d], w);
}

// ---- pass 3: weighted scatter-add of transformed src features ----
__global__ void k_agg(const float* __restrict__ WE, const float* __restrict__ denom,
                      const int* __restrict__ dst, const int* __restrict__ src,
                      const int* __restrict__ etype, const float* __restrict__ Y,
                      float* __restrict__ acc, int out_dim) {
    int e = blockIdx.x * blockDim.x + threadIdx.x;
    if (e >= N_EDGES) return;
    const int s = src[e], d = dst[e], r = etype[e];
    const float coeff = WE[e] / (denom[d] + 1e-16f);
    const float* ys = Y + ((size_t)r * N_NODES + s) * out_dim;
    float* ad = acc + (size_t)d * out_dim;
    for (int c = 0; c < out_dim; ++c) atomicAdd(&ad[c], coeff * ys[c]);
}

// ---- bias + optional leaky_relu(0.01) ----
__global__ void k_finish(const float* __restrict__ acc, const float* __restrict__ b,
                         float* __restrict__ out, int out_dim, int apply_act) {
    int i = blockIdx.x * blockDim.x + threadIdx.x;
    if (i >= N_NODES * out_dim) return;
    float v = acc[i] + b[i % out_dim];
    if (apply_act) v = (v > 0.f) ? v : 0.01f * v;
    out[i] = v;
}

// ---- s_ = H2 @ H2^T (8192x8192, K=32) via f32 WMMA ----
// One wave computes a 16x64 strip: the K=32 A-slice is loaded once (8x b64)
// and reused across 4 column tiles (32 v_wmma per wave). Output rows span
// 256 B contiguous across the j-loop.
__global__ __launch_bounds__(32)
void k_simmat(const float* __restrict__ H, float* __restrict__ S) {
    const int lane = threadIdx.x;
    const int m   = lane & 15;
    const int kh  = lane >> 4;
    const int ib  = blockIdx.x * 16;
    const int jb0 = blockIdx.y * 64;

    v2f a[8];
#pragma unroll
    for (int t = 0; t < 8; ++t) {
        const int ka = t * 4 + kh * 2;
        a[t] = *(const v2f*)&H[(size_t)(ib + m) * 32 + ka];
    }

#pragma unroll
    for (int j = 0; j < 4; ++j) {
        const int jb = jb0 + j * 16;
        v8f c = {0.f, 0.f, 0.f, 0.f, 0.f, 0.f, 0.f, 0.f};
#pragma unroll
        for (int t = 0; t < 8; ++t) {
            const int ka = t * 4 + kh * 2;
            v2f b = *(const v2f*)&H[(size_t)(jb + m) * 32 + ka]; // B[k][n]=H[jb+n][k]
            c = __builtin_amdgcn_wmma_f32_16x16x4_f32(false, a[t], false, b,
                                                      (short)0, c, false, false);
        }
        const int mbase = kh * 8;
#pragma unroll
        for (int v = 0; v < 8; ++v)
            S[(size_t)(ib + mbase + v) * N_NODES + jb + m] = c[v];
    }
}

extern "C" void kernel_launch(void* const* d_in, const int* in_sizes, int n_in,
                              void* d_out_, int out_size, void* d_ws, size_t ws_size,
                              hipStream_t stream) {
    const int*   x     = (const int*)d_in[0];
    const int*   eidx  = (const int*)d_in[1];
    const float* eattr = (const float*)d_in[2];
    const int*   etype = (const int*)d_in[3];
    const float* emb   = (const float*)d_in[4];
    const float* W1 = (const float*)d_in[5];
    const float* q1 = (const float*)d_in[6];
    const float* k1 = (const float*)d_in[7];
    const float* e1 = (const float*)d_in[8];
    const float* le1 = (const float*)d_in[9];
    const float* b1 = (const float*)d_in[10];
    const float* W2 = (const float*)d_in[11];
    const float* q2 = (const float*)d_in[12];
    const float* k2 = (const float*)d_in[13];
    const float* e2 = (const float*)d_in[14];
    const float* le2 = (const float*)d_in[15];
    const float* b2 = (const float*)d_in[16];
    const float* W3 = (const float*)d_in[17];
    const float* q3 = (const float*)d_in[18];
    const float* k3 = (const float*)d_in[19];
    const float* e3 = (const float*)d_in[20];
    const float* le3 = (const float*)d_in[21];
    const float* b3 = (const float*)d_in[22];

    float* dout = (float*)d_out_;
    const int* src = eidx;             // edge_index[0]
    const int* dst = eidx + N_EDGES;   // edge_index[1]

    // workspace layout (floats); total ~17 MB
    float* ws = (float*)d_ws;
    size_t off = 0;
    float* X0 = ws + off; off += (size_t)N_NODES * EMBED;
    float* Y  = ws + off; off += (size_t)R_REL * N_NODES * 32;
    float* AI = ws + off; off += (size_t)R_REL * N_NODES;
    float* AJ = ws + off; off += (size_t)R_REL * N_NODES;
    unsigned int* AMAX = (unsigned int*)(ws + off); off += N_NODES;
    float* DEN = ws + off; off += N_NODES;
    float* WE  = ws + off; off += N_EDGES;
    float* ACC = ws + off; off += (size_t)N_NODES * 32;
    float* H   = ws + off; off += (size_t)N_NODES * 16;
    float* H2  = ws + off; off += (size_t)N_NODES * 32;
    float* LEV = ws + off; off += 32;

    float* outEmb = dout;
    float* outX   = dout + (size_t)N_NODES * EMBED;
    float* outS   = outX + (size_t)N_NODES * 32;

    { int t = N_NODES * EMBED;
      k_embed<<<(t + 255) / 256, 256, 0, stream>>>(x, emb, X0, outEmb); }

    // everything after the per-relation GEMM is dimension-generic
    auto run_attention = [&](int out_dim, const float* q, const float* kk,
                             const float* ev, const float* le, const float* b,
                             float* outbuf, int act) {
        k_lev<<<1, EDGE_DIM, 0, stream>>>(le, ev, LEV, out_dim);
        { int t = R_REL * N_NODES;
          k_scores<<<(t + 255) / 256, 256, 0, stream>>>(Y, q, kk, AI, AJ, out_dim); }
        { int t = N_NODES * 32;
          k_init<<<(t + 255) / 256, 256, 0, stream>>>(AMAX, DEN, ACC, N_NODES * out_dim); }
        { int t = N_EDGES;
          k_alpha<<<(t + 255) / 256, 256, 0, stream>>>(src, dst, etype, eattr, LEV,
                                                       AI, AJ, WE, AMAX); }
        { int t = N_EDGES;
          k_expsum<<<(t + 255) / 256, 256, 0, stream>>>(WE, AMAX, dst, DEN); }
        { int t = N_EDGES;
          k_agg<<<(t + 255) / 256, 256, 0, stream>>>(WE, DEN, dst, src, etype, Y,
                                                     ACC, out_dim); }
        { int t = N_NODES * out_dim;
          k_finish<<<(t + 255) / 256, 256, 0, stream>>>(ACC, b, outbuf, out_dim, act); }
    };

    // layer 1: 128 -> 16, leaky_relu(0.01) into H
    {
        dim3 g(N_NODES / 16, 1, R_REL);
        k_gemm_x_w<EMBED, 16><<<g, 32, 0, stream>>>(X0, W1, Y);
        run_attention(16, q1, k1, e1, le1, b1, H, 1);
    }
    // layer 2: 16 -> 32 into H2 (feeds s_ = H2 @ H2^T)
    {
        dim3 g(N_NODES / 16, 2, R_REL);
        k_gemm_x_w<16, 32><<<g, 32, 0, stream>>>(H, W2, Y);
        run_attention(32, q2, k2, e2, le2, b2, H2, 0);
    }
    // layer 3: 16 -> 32 directly into x_ output
    {
        dim3 g(N_NODES / 16, 2, R_REL);
        k_gemm_x_w<16, 32><<<g, 32, 0, stream>>>(H, W3, Y);
        run_attention(32, q3, k3, e3, le3, b3, outX, 0);
    }

    dim3 gs(N_NODES / 16, N_NODES / 64);
    k_simmat<<<gs, 32, 0, stream>>>(H2, outS);
}